// RelativePositionalEncoding_9070970929735
// MI455X (gfx1250) — compile-verified
//
#include <hip/hip_runtime.h>
#include <stdint.h>

// Problem constants (fixed by the reference's setup_inputs).
#define N_TOK   2048      // N
#define N_REL   65        // 2*32+1 relative-position bins
#define ADIM    8         // attn_dim
#define MAXREL  32

typedef float    v4f  __attribute__((ext_vector_type(4)));
typedef uint32_t v4u  __attribute__((ext_vector_type(4)));
typedef int      v8i  __attribute__((ext_vector_type(8)));
typedef int      v4i  __attribute__((ext_vector_type(4)));

// Issue a TDM DMA of `n_elems` 4-byte elements from global `gptr` into LDS at
// byte offset `lds_byte_addr`. Descriptor bit layout per CDNA5 ISA §8
// (Tensor DMA Descriptor): group0 = {count/type, lds_addr, global_addr},
// group1 = {data_size, tensor_dim0/1, tile_dim0/1/2, strides}. 1-D tile:
// dim0 = tile0 = n, dim1 = tile1 = 1, tile2 = 0, stride0 = n.
// This toolchain's builtin takes 6 args (amdgpu-toolchain / clang-23 form):
//   (uint32x4 g0, int32x8 g1, int32x4 g2, int32x4 g3, int32x8 gx, i32 cpol)
__device__ __forceinline__ void tdm_load_1d_f32(uint32_t lds_byte_addr,
                                                const void* gptr,
                                                uint32_t n_elems) {
    uint64_t ga = (uint64_t)(uintptr_t)gptr;

    v4u g0;
    g0.x = 1u;                                              // count=1 (valid), user mode
    g0.y = lds_byte_addr;                                   // lds_addr[31:0]
    g0.z = (uint32_t)(ga & 0xFFFFFFFFu);                    // global_addr[31:0]
    g0.w = ((uint32_t)(ga >> 32) & 0x01FFFFFFu)             // global_addr[56:32]
         | (2u << 30);                                      // type = 2 ("image")

    const uint32_t dim0 = n_elems, dim1 = 1u;
    const uint32_t tile0 = n_elems, tile1 = 1u, tile2 = 0u;
    const uint64_t s0 = n_elems;                            // tensor_dim0_stride
    const uint64_t s1 = 0u;                                 // tensor_dim1_stride

    v8i g1;
    g1[0] = (int)(2u << 16);                                // data_size=2 -> 4 bytes
    g1[1] = (int)((dim0 & 0xFFFFu) << 16);                  // tensor_dim0[15:0]
    g1[2] = (int)(((dim0 >> 16) & 0xFFFFu)                  // tensor_dim0[31:16]
                | ((dim1 & 0xFFFFu) << 16));                // tensor_dim1[15:0]
    g1[3] = (int)(((dim1 >> 16) & 0xFFFFu)                  // tensor_dim1[31:16]
                | ((tile0 & 0xFFFFu) << 16));               // tile_dim0
    g1[4] = (int)((tile1 & 0xFFFFu) | ((tile2 & 0xFFFFu) << 16));
    g1[5] = (int)(uint32_t)(s0 & 0xFFFFFFFFu);              // stride0[31:0]
    g1[6] = (int)(((uint32_t)(s0 >> 32) & 0xFFFFu)          // stride0[47:32]
                | (((uint32_t)s1 & 0xFFFFu) << 16));        // stride1[15:0]
    g1[7] = (int)(uint32_t)((s1 >> 16) & 0xFFFFFFFFu);      // stride1[47:16]

    v4i z4 = {0, 0, 0, 0};                                  // groups 2/3 unused (<=2D)
    v8i z8 = {0, 0, 0, 0, 0, 0, 0, 0};                      // extra group (unused)
    __builtin_amdgcn_tensor_load_to_lds(g0, g1, z4, z4, z8, /*cpol=*/0);
}

__global__ void __launch_bounds__(256)
relpos_kernel(const int* __restrict__ index,   // (B*N,)
              const float* __restrict__ W,     // (65, 8)
              const float* __restrict__ b,     // (8,)
              float* __restrict__ out)         // (B*N, N, 8)
{
    __shared__ float tab[N_REL * ADIM];        // 520 f32 = 2080 B (16B-aligned size)
    __shared__ int   jrow[N_TOK];              // 2048 i32 = 8 KB

    const int row  = blockIdx.x;               // flat (b, i) row, 0 .. B*N-1
    const int bidx = row >> 11;                // row / N_TOK
    const int tid  = threadIdx.x;

    // Wave 0 stages the W table and this batch's index row into LDS via the
    // Tensor Data Mover, then drains TENSORcnt before releasing the barrier.
    if (tid < 32) {
        tdm_load_1d_f32((uint32_t)(uintptr_t)(void*)tab,  W, N_REL * ADIM);
        tdm_load_1d_f32((uint32_t)(uintptr_t)(void*)jrow,
                        index + (size_t)bidx * N_TOK, N_TOK);
        __builtin_amdgcn_s_wait_tensorcnt(0);
    }
    __syncthreads();

    const int idx_i = index[row];              // block-uniform -> scalar load

    // Each thread has a fixed float4 parity: g = tid + 256k, 256 is even.
    const int h = tid & 1;                     // which half of the 8-dim vector
    const v4f bh = *(const v4f*)(b + 4 * h);   // bias half, kept in VGPRs

    const v4f* tab4 = (const v4f*)tab;
    float* orow = out + (size_t)row * (size_t)(N_TOK * ADIM);

    // N_TOK * ADIM / 4 = 4096 float4 groups per row; 256 threads -> 16 iters.
    // Per wave-iteration: 512 B fully-coalesced nontemporal store stream.
    #pragma unroll 4
    for (int k = 0; k < (N_TOK * ADIM / 4) / 256; ++k) {
        const int g = tid + (k << 8);
        const int j = g >> 1;
        int d = idx_i - jrow[j];
        d = d < -MAXREL ? -MAXREL : (d > MAXREL ? MAXREL : d);
        const int c = d + MAXREL;              // 0 .. 64
        const v4f t = tab4[c * 2 + h];         // ds_load_b128 gather
        const v4f r = t + bh;
        __builtin_nontemporal_store(r, (v4f*)(orow + (size_t)g * 4));
    }
}

extern "C" void kernel_launch(void* const* d_in, const int* in_sizes, int n_in,
                              void* d_out, int out_size, void* d_ws, size_t ws_size,
                              hipStream_t stream) {
    const int*   index = (const int*)d_in[0];   // (B, N) int32
    const float* W     = (const float*)d_in[1]; // (65, 8) f32
    const float* b     = (const float*)d_in[2]; // (8,) f32
    float*       out   = (float*)d_out;         // (B, N, N, 8) f32

    const int rows = in_sizes[0];               // B*N = 4096
    relpos_kernel<<<rows, 256, 0, stream>>>(index, W, b, out);
}